// GCN_90288802497367
// MI455X (gfx1250) — compile-verified
//
#include <hip/hip_runtime.h>
#include <hip/hip_bf16.h>

typedef __attribute__((ext_vector_type(2))) float v2f;
typedef __attribute__((ext_vector_type(8))) float v8f;

#define F_IN   128
#define F_HID  16

// ---- fast non-returning f32 atomic add (global_atomic_add_f32) -------------
__device__ __forceinline__ void atomic_add_f32(float* p, float v) {
#if defined(__AMDGCN__)
    unsafeAtomicAdd(p, v);
#else
    atomicAdd(p, v);
#endif
}

// ---------------------------------------------------------------------------
// 1) deg[i] = 1.0f  (self-loop contribution)
__global__ void gcn_init_deg(float* __restrict__ deg, int n) {
    int i = blockIdx.x * blockDim.x + threadIdx.x;
    if (i < n) deg[i] = 1.0f;
}

// 2) deg[dst] += 1 per edge
__global__ void gcn_deg_accum(const int* __restrict__ ei, float* __restrict__ deg,
                              long long E) {
    long long e = (long long)blockIdx.x * blockDim.x + threadIdx.x;
    if (e >= E) return;
    int d = ei[E + e];                       // row 1 of edge_index = dst
    atomic_add_f32(&deg[d], 1.0f);
}

// 3) dis[i] = rsqrt(deg[i])  (deg >= 1 always; in place)
__global__ void gcn_rsqrt(float* __restrict__ deg_dis, int n) {
    int i = blockIdx.x * blockDim.x + threadIdx.x;
    if (i < n) {
        float dgv = deg_dis[i];
        deg_dis[i] = dgv > 0.0f ? rsqrtf(dgv) : 0.0f;
    }
}

// ---------------------------------------------------------------------------
// 4) xW = x @ W1  via V_WMMA_F32_16X16X4_F32, one wave per 16-row tile.
//    A (16x4 f32): lanes 0-15 -> {K=k0,k0+1}, lanes 16-31 -> {K=k0+2,k0+3}, M=lane%16
//    B (4x16 f32): lanes index N, VGPR pair holds the matching K pair (assumed
//    symmetric with A per ISA "row striped across lanes" rule)
//    C/D (16x16 f32): VGPR r -> row r (lanes 0-15) / row r+8 (lanes 16-31), col = lane%16
__global__ __launch_bounds__(32)
void gcn_xw_wmma(const float* __restrict__ x, const float* __restrict__ W1,
                 float* __restrict__ xW, int n) {
    const int row0 = blockIdx.x * 16;
    if (row0 >= n) return;
    const int lane = threadIdx.x;        // 0..31
    const int half = lane >> 4;          // 0 | 1
    const int l16  = lane & 15;

    const float* xrow = x + (long long)(row0 + l16) * F_IN;

    // prefetch this lane's row of the next tile (global_prefetch_b8)
    if (row0 + 16 + l16 < n)
        __builtin_prefetch(x + (long long)(row0 + 16 + l16) * F_IN + half * 64, 0, 3);

    v8f c = {};
#pragma unroll
    for (int k0 = 0; k0 < F_IN; k0 += 4) {
        const int ka = k0 + half * 2;
        v2f a, b;
        a.x = xrow[ka];
        a.y = xrow[ka + 1];
        b.x = W1[ka * F_HID + l16];
        b.y = W1[(ka + 1) * F_HID + l16];
        c = __builtin_amdgcn_wmma_f32_16x16x4_f32(
                /*neg_a=*/false, a, /*neg_b=*/false, b,
                /*c_mod=*/(short)0, c, /*reuse_a=*/false, /*reuse_b=*/false);
    }
#pragma unroll
    for (int r = 0; r < 8; ++r) {
        const int m = r + half * 8;
        xW[(long long)(row0 + m) * F_HID + l16] = c[r];
    }
}

// ---------------------------------------------------------------------------
// 5) h1agg[i][j] = xW[i][j] * dis[i]^2   (self-loop seed)
__global__ void gcn_agg1_init(const float* __restrict__ xW, const float* __restrict__ dis,
                              float* __restrict__ h1agg, int n) {
    long long t = (long long)blockIdx.x * blockDim.x + threadIdx.x;
    long long i = t >> 4;
    if (i >= n) return;
    float dv = dis[i];
    h1agg[t] = xW[t] * dv * dv;
}

// 6) per (edge, j): h1agg[dst][j] += xW[src][j] * dis[src]*dis[dst]
__global__ void gcn_agg1_edges(const int* __restrict__ ei, const float* __restrict__ dis,
                               const float* __restrict__ xW, float* __restrict__ h1agg,
                               long long E) {
    long long t = (long long)blockIdx.x * blockDim.x + threadIdx.x;
    long long e = t >> 4;
    int j = (int)(t & 15);
    if (e >= E) return;
    int s = ei[e];
    int d = ei[E + e];
    float norm = dis[s] * dis[d];           // half-wave-uniform loads, L2-resident
    atomic_add_f32(&h1agg[(long long)d * F_HID + j],
                   xW[(long long)s * F_HID + j] * norm);
}

// 7) z[i] = relu(h1agg[i] + b1) . W2
__global__ void gcn_l1_epilogue(const float* __restrict__ h1agg,
                                const float* __restrict__ b1,
                                const float* __restrict__ W2,
                                float* __restrict__ z, int n) {
    int i = blockIdx.x * blockDim.x + threadIdx.x;
    if (i >= n) return;
    const float* row = h1agg + (long long)i * F_HID;
    float acc = 0.0f;
#pragma unroll
    for (int j = 0; j < F_HID; ++j) {
        float hv = row[j] + b1[j];
        hv = fmaxf(hv, 0.0f);
        acc = fmaf(hv, W2[j], acc);
    }
    z[i] = acc;
}

// 8) out[i] = b2 + z[i]*dis[i]^2   (bias + self-loop seed)
__global__ void gcn_out_init(const float* __restrict__ z, const float* __restrict__ dis,
                             const float* __restrict__ b2, float* __restrict__ out, int n) {
    int i = blockIdx.x * blockDim.x + threadIdx.x;
    if (i >= n) return;
    float dv = dis[i];
    out[i] = b2[0] + z[i] * dv * dv;
}

// 9) per edge: out[dst] += z[src] * dis[src]*dis[dst]
__global__ void gcn_agg2_edges(const int* __restrict__ ei, const float* __restrict__ dis,
                               const float* __restrict__ z, float* __restrict__ out,
                               long long E) {
    long long e = (long long)blockIdx.x * blockDim.x + threadIdx.x;
    if (e >= E) return;
    int s = ei[e];
    int d = ei[E + e];
    atomic_add_f32(&out[d], z[s] * dis[s] * dis[d]);
}

// ---------------------------------------------------------------------------
extern "C" void kernel_launch(void* const* d_in, const int* in_sizes, int n_in,
                              void* d_out, int out_size, void* d_ws, size_t ws_size,
                              hipStream_t stream) {
    const float* x  = (const float*)d_in[0];
    const int*   ei = (const int*)d_in[1];     // edge_index, shape (2,E) row-major
    const float* W1 = (const float*)d_in[2];
    const float* b1 = (const float*)d_in[3];
    const float* W2 = (const float*)d_in[4];
    const float* b2 = (const float*)d_in[5];
    float* out = (float*)d_out;

    const int       N = in_sizes[0] / F_IN;    // 100000
    const long long E = (long long)in_sizes[1] / 2;

    // workspace layout (floats): [deg/dis : N][xW : 16N][h1agg : 16N][z : N]
    float* ws  = (float*)d_ws;
    float* dis = ws;                           // deg, then rsqrt in place
    float* xW  = ws + N;
    float* h1a = ws + (long long)17 * N;
    float* z   = ws + (long long)33 * N;

    const int T = 256;
    const unsigned nb_n    = (unsigned)((N + T - 1) / T);
    const unsigned nb_n16  = (unsigned)(((long long)N * F_HID + T - 1) / T);
    const unsigned nb_e    = (unsigned)((E + T - 1) / T);
    const unsigned nb_e16  = (unsigned)((E * F_HID + T - 1) / T);
    const unsigned nb_tile = (unsigned)((N + 15) / 16);

    gcn_init_deg   <<<nb_n,    T,  0, stream>>>(dis, N);
    gcn_deg_accum  <<<nb_e,    T,  0, stream>>>(ei, dis, E);
    gcn_rsqrt      <<<nb_n,    T,  0, stream>>>(dis, N);
    gcn_xw_wmma    <<<nb_tile, 32, 0, stream>>>(x, W1, xW, N);
    gcn_agg1_init  <<<nb_n16,  T,  0, stream>>>(xW, dis, h1a, N);
    gcn_agg1_edges <<<nb_e16,  T,  0, stream>>>(ei, dis, xW, h1a, E);
    gcn_l1_epilogue<<<nb_n,    T,  0, stream>>>(h1a, b1, W2, z, N);
    gcn_out_init   <<<nb_n,    T,  0, stream>>>(z, dis, b2, out, N);
    gcn_agg2_edges <<<nb_e,    T,  0, stream>>>(ei, dis, z, out, E);
}